// SessionLinearBank_52493090292188
// MI455X (gfx1250) — compile-verified
//
#include <hip/hip_runtime.h>
#include <hip/hip_bf16.h>

typedef __attribute__((ext_vector_type(2))) float v2f;
typedef __attribute__((ext_vector_type(8))) float v8f;

#define BDIM 4096
#define DIM  1024
#define NEXP 9
#define KC   64          // K chunk staged in LDS
#define LDP  68          // padded LDS row stride (floats): conflict-free b64 frag loads
#define BLKN 64          // cols per block (4 waves x 16)

// ---------------- binning kernels ----------------

__global__ void sess_zero_meta(int* __restrict__ meta) {
    if (threadIdx.x < 48) meta[threadIdx.x] = 0;   // counts[16] | offsets[16] | cursor[16]
}

__global__ void sess_hist(const int* __restrict__ idx, int* __restrict__ counts, int n) {
    int i = blockIdx.x * blockDim.x + threadIdx.x;
    if (i < n) atomicAdd(&counts[idx[i]], 1);
}

__global__ void sess_scan(const int* __restrict__ counts,
                          int* __restrict__ offsets, int* __restrict__ cursor) {
    if (threadIdx.x == 0) {
        int run = 0;
        for (int e = 0; e < NEXP; ++e) { offsets[e] = run; cursor[e] = run; run += counts[e]; }
    }
}

__global__ void sess_scatter(const int* __restrict__ idx, int* __restrict__ cursor,
                             int* __restrict__ perm, int n) {
    int i = blockIdx.x * blockDim.x + threadIdx.x;
    if (i < n) { int pos = atomicAdd(&cursor[idx[i]], 1); perm[pos] = i; }
}

// ---------------- WMMA GEMM per expert bin ----------------
// block: 128 threads (4 waves). tile: 16 rows x 64 cols. K loop in 64-chunks via LDS.

__global__ __launch_bounds__(128)
void sess_gemm(const float* __restrict__ x, const float* __restrict__ W,
               const float* __restrict__ bias,
               const int* __restrict__ counts, const int* __restrict__ offsets,
               const int* __restrict__ perm, float* __restrict__ out) {
    const int e        = blockIdx.z;
    const int cnt      = counts[e];
    const int row_tile = blockIdx.y;
    if (row_tile * 16 >= cnt) return;            // uniform early exit (EXEC stays full)
    const int base = offsets[e];
    const int nb   = blockIdx.x * BLKN;

    __shared__ float Alds[16 * LDP];
    __shared__ float Blds[BLKN * LDP];
    __shared__ int   rows[16];

    const int tid  = threadIdx.x;
    const int wave = tid >> 5;
    const int lane = tid & 31;
    const int nl   = lane & 15;                  // N (and A-frag M) within 16
    const int hi   = lane >> 4;                  // K sub-offset selector

    if (tid < 16) {
        int k  = row_tile * 16 + tid;
        int kc = (k < cnt) ? k : (cnt - 1);      // clamp tail (stores masked later)
        rows[tid] = perm[base + kc];
    }
    __syncthreads();

    // C init = bias[e][n] broadcast across M
    const int n_wave = nb + wave * 16;
    const float bv = bias[e * DIM + n_wave + nl];
    v8f acc;
#pragma unroll
    for (int i = 0; i < 8; ++i) acc[i] = bv;

    const float* __restrict__ Wexp = W + (size_t)e * DIM * DIM;

    for (int k0 = 0; k0 < DIM; k0 += KC) {
        __syncthreads();
        // A tile: 16 rows x 64 floats = 256 float4, 2 per thread (gathered rows)
        {
            int f4 = tid;
#pragma unroll
            for (int it = 0; it < 2; ++it, f4 += 128) {
                int r = f4 >> 4, c4 = f4 & 15;
                const float4 v = *(const float4*)(x + (size_t)rows[r] * DIM + k0 + c4 * 4);
                float* dst = &Alds[r * LDP + c4 * 4];
                dst[0] = v.x; dst[1] = v.y; dst[2] = v.z; dst[3] = v.w;
            }
        }
        // B tile: W[e][nb..nb+63][k0..k0+63] stored [n][k] = 1024 float4, 8 per thread
        {
            int f4 = tid;
#pragma unroll
            for (int it = 0; it < 8; ++it, f4 += 128) {
                int n = f4 >> 4, c4 = f4 & 15;
                const float4 v = *(const float4*)(Wexp + (size_t)(nb + n) * DIM + k0 + c4 * 4);
                float* dst = &Blds[n * LDP + c4 * 4];
                dst[0] = v.x; dst[1] = v.y; dst[2] = v.z; dst[3] = v.w;
            }
        }
        __syncthreads();

        // 16 x V_WMMA_F32_16X16X4_F32 per chunk
        const float* __restrict__ Arow = &Alds[nl * LDP];                 // A-frag M = lane&15
        const float* __restrict__ Brow = &Blds[(wave * 16 + nl) * LDP];   // B col n, W is [n][k]
#pragma unroll
        for (int kk = 0; kk < KC; kk += 4) {
            v2f a, bb;
            a.x  = Arow[kk + hi * 2 + 0];   // lanes 0-15: K=0,1 ; lanes 16-31: K=2,3
            a.y  = Arow[kk + hi * 2 + 1];
            bb.x = Brow[kk + hi * 2 + 0];   // B[k][n] = W[n][k], same striping
            bb.y = Brow[kk + hi * 2 + 1];
            acc = __builtin_amdgcn_wmma_f32_16x16x4_f32(
                false, a, false, bb, (short)0, acc, false, false);
        }
    }

    // scatter-store: VGPR v -> M = v + 8*hi, N = n_wave + nl
#pragma unroll
    for (int v = 0; v < 8; ++v) {
        int m = v + 8 * hi;
        if (row_tile * 16 + m < cnt) {
            out[(size_t)rows[m] * DIM + n_wave + nl] = acc[v];
        }
    }
}

// ---------------- launcher ----------------

extern "C" void kernel_launch(void* const* d_in, const int* in_sizes, int n_in,
                              void* d_out, int out_size, void* d_ws, size_t ws_size,
                              hipStream_t stream) {
    const float* x    = (const float*)d_in[0];
    const float* W    = (const float*)d_in[1];
    const float* bias = (const float*)d_in[2];
    const int*   sidx = (const int*)d_in[3];
    float*       out  = (float*)d_out;

    int* meta    = (int*)d_ws;
    int* counts  = meta + 0;
    int* offsets = meta + 16;
    int* cursor  = meta + 32;
    int* perm    = meta + 48;

    sess_zero_meta<<<1, 64, 0, stream>>>(meta);
    sess_hist<<<(BDIM + 255) / 256, 256, 0, stream>>>(sidx, counts, BDIM);
    sess_scan<<<1, 32, 0, stream>>>(counts, offsets, cursor);
    sess_scatter<<<(BDIM + 255) / 256, 256, 0, stream>>>(sidx, cursor, perm, BDIM);

    dim3 grid(DIM / BLKN, (BDIM + 15) / 16, NEXP);   // 16 x 256 x 9, most exit early
    sess_gemm<<<grid, 128, 0, stream>>>(x, W, bias, counts, offsets, perm, out);
}